// PiecewiseLinearBasis_63479616635238
// MI455X (gfx1250) — compile-verified
//
#include <hip/hip_runtime.h>

// Piecewise-linear (hat) basis expansion on [-1, 1].
// out[i, k] = (1 - frac_i) * (k == L_i)  +  frac_i * (k == R_i)
// N = 1M rows, K = 128 knots -> 512 MiB of output stores per call.
// Pure store-bandwidth-bound (~23 us floor @ 23.3 TB/s). One wave32 writes
// one 128-float row as 32 lanes x float4 non-temporal b128 stores (512 B/row).

#define PLB_DOMAIN_MIN (-1.0f)
#define PLB_DOMAIN_MAX ( 1.0f)

typedef float v4f __attribute__((ext_vector_type(4)));

__device__ __forceinline__ float plb_bcast_lane(float v, int lane) {
  return __int_as_float(__builtin_amdgcn_readlane(__float_as_int(v), lane));
}

// Fast path: K == 128. One wave handles 32 consecutive rows.
//  - coalesced b32 load of 32 inputs (one per lane)
//  - clamp + division done ONCE per lane (vector), before broadcast
//  - v_readlane broadcast of each row's pre-scaled coordinate
//  - each lane computes its 4 columns and issues one NT b128 store
__global__ __launch_bounds__(256) void plb_k128_kernel(
    const float* __restrict__ in, float* __restrict__ out, int n, float step) {
  const int lane   = threadIdx.x & 31;
  const int waveId = (blockIdx.x * 256 + threadIdx.x) >> 5;
  const int base   = waveId << 5;            // 32 rows per wave
  if (base >= n) return;

  const int r = base + lane;
  float x = (r < n) ? in[r] : 0.0f;          // coalesced; 4 MiB total input
  // Hoisted per-lane preprocessing: one clamp + one divide for 32 rows.
  x = fminf(fmaxf(x, PLB_DOMAIN_MIN), PLB_DOMAIN_MAX);
  const float scaled_v = (x - PLB_DOMAIN_MIN) / step;   // in [0, 127]

  const int rows = (n - base) < 32 ? (n - base) : 32;
  const int k0   = lane << 2;                // this lane's first column

  for (int i = 0; i < rows; ++i) {
    const float scaled = plb_bcast_lane(scaled_v, i);    // sgpr broadcast
    int li = (int)scaled;                                // trunc == floor (>=0)
    li = li < 127 ? li : 127;
    const int ri = (li + 1) < 127 ? (li + 1) : 127;
    const float frac = scaled - (float)li;               // == reference fraction
    const float omf  = 1.0f - frac;

    v4f v;
#pragma unroll
    for (int j = 0; j < 4; ++j) {
      const int k = k0 + j;
      float val = (k == li) ? omf : 0.0f;
      if (k == ri) val += frac;      // li==ri==127 -> omf+frac == 1.0 exactly
      v[j] = val;
    }
    // 512 MiB stream >> 192 MB L2: non-temporal to avoid L2 thrash.
    __builtin_nontemporal_store(
        v, (v4f*)(out + (size_t)(base + i) * 128 + (size_t)k0));
  }
}

// Generic fallback for K != 128 (elementwise over the output).
__global__ __launch_bounds__(256) void plb_generic_kernel(
    const float* __restrict__ in, float* __restrict__ out,
    int K, float step, long long total) {
  long long idx          = (long long)blockIdx.x * 256 + threadIdx.x;
  const long long stride = (long long)gridDim.x * 256;
  const int kmax = K - 1;
  for (; idx < total; idx += stride) {
    const int row = (int)(idx / K);
    const int k   = (int)(idx % K);
    float x = in[row];
    x = fminf(fmaxf(x, PLB_DOMAIN_MIN), PLB_DOMAIN_MAX);
    const float scaled = (x - PLB_DOMAIN_MIN) / step;
    int li = (int)scaled;
    li = li < kmax ? li : kmax;
    const int ri = (li + 1) < kmax ? (li + 1) : kmax;
    const float frac = scaled - (float)li;
    float val = (k == li) ? (1.0f - frac) : 0.0f;
    if (k == ri) val += frac;
    __builtin_nontemporal_store(val, out + idx);
  }
}

extern "C" void kernel_launch(void* const* d_in, const int* in_sizes, int n_in,
                              void* d_out, int out_size, void* d_ws, size_t ws_size,
                              hipStream_t stream) {
  (void)n_in; (void)d_ws; (void)ws_size;
  const float* in = (const float*)d_in[0];
  float* out      = (float*)d_out;
  const int n     = in_sizes[0];
  if (n <= 0 || out_size <= 0) return;

  const int K = out_size / n;  // num_knots (128 for the reference setup)
  const float step =
      (K > 1) ? (PLB_DOMAIN_MAX - PLB_DOMAIN_MIN) / (float)(K - 1) : 1.0f;

  if (K == 128) {
    const int waves  = (n + 31) / 32;        // one wave per 32 rows
    const int blocks = (waves + 7) / 8;      // 8 waves per 256-thread block
    plb_k128_kernel<<<blocks, 256, 0, stream>>>(in, out, n, step);
  } else {
    const long long total = (long long)out_size;
    long long b = (total + 255) / 256;
    if (b > 262144) b = 262144;              // grid-stride cap
    plb_generic_kernel<<<(int)b, 256, 0, stream>>>(in, out, K, step, total);
  }
}